// AttentionLayer_s_27281632264369
// MI455X (gfx1250) — compile-verified
//
#include <hip/hip_runtime.h>
#include <hip/hip_bf16.h>

typedef __attribute__((ext_vector_type(2))) float v2f;
typedef __attribute__((ext_vector_type(8))) float v8f;

#define NEG_INF (-__builtin_inff())

// Problem constants (from reference)
constexpr int Bq   = 16;
constexpr int T    = 12;
constexpr int Nn   = 1024;
constexpr int D    = 128;
constexpr int H    = 8;
constexpr int HD   = 16;          // head dim
constexpr int TOPK = 50;
constexpr int Mm   = 20;          // memory nodes
constexpr int HB   = H * Bq;      // 128
constexpr int ROWS = Bq * T * Nn; // 196608

constexpr size_t QN    = (size_t)HB * T * Nn * HD; // 25,165,824 floats per split-head tensor
constexpr size_t CNT_N = (size_t)HB * T * Nn;      // 1,572,864
constexpr size_t IDX_N = (size_t)HB * T * Mm * TOPK;

static __device__ __forceinline__ v8f wmma4(v2f a, v2f b, v8f c) {
    return __builtin_amdgcn_wmma_f32_16x16x4_f32(false, a, false, b, (short)0, c, false, false);
}

// ---------------------------------------------------------------------------
// 0) init workspace: agg = 0, cnt = EPS
// ---------------------------------------------------------------------------
__global__ void init_ws(float* __restrict__ agg, float* __restrict__ cnt) {
    size_t i = (size_t)blockIdx.x * blockDim.x + threadIdx.x;
    if (i < QN)    agg[i] = 0.0f;
    if (i < CNT_N) cnt[i] = 1e-14f;
}

// ---------------------------------------------------------------------------
// 1) q/k/v projections: X(196608x128) @ W(128x128) + b, written in split-head
//    layout (HB,T,N,16). grid=(ROWS/16, 3), block=256 (8 waves = 8 col tiles)
// ---------------------------------------------------------------------------
__global__ __launch_bounds__(256) void qkv_proj(
    const float* __restrict__ q_in, const float* __restrict__ k_in,
    const float* __restrict__ v_in,
    const float* __restrict__ Wq, const float* __restrict__ bq,
    const float* __restrict__ Wk, const float* __restrict__ bk,
    const float* __restrict__ Wv, const float* __restrict__ bv,
    float* __restrict__ qs, float* __restrict__ ks, float* __restrict__ vs)
{
    const int z = blockIdx.y;
    const float* X    = (z == 0) ? q_in : (z == 1) ? k_in : v_in;
    const float* W    = (z == 0) ? Wq   : (z == 1) ? Wk   : Wv;
    const float* bias = (z == 0) ? bq   : (z == 1) ? bk   : bv;
    float*       OUT  = (z == 0) ? qs   : (z == 1) ? ks   : vs;

    const int wave = threadIdx.x >> 5, lane = threadIdx.x & 31;
    const int half = lane >> 4, l16 = lane & 15;
    const int row0 = blockIdx.x * 16;
    const int col0 = wave * 16;            // wave == head index (D/16 == 8)
    const int arow = row0 + l16;

    v8f acc = {0.f,0.f,0.f,0.f,0.f,0.f,0.f,0.f};
    for (int k = 0; k < D; k += 4) {
        const int ka = k + 2 * half;
        if (k + 4 < D) __builtin_prefetch(&W[(k + 4) * D + col0 + l16], 0, 1);
        v2f a; a[0] = X[(size_t)arow * D + ka];       a[1] = X[(size_t)arow * D + ka + 1];
        v2f b; b[0] = W[(size_t)ka * D + col0 + l16]; b[1] = W[(size_t)(ka + 1) * D + col0 + l16];
        acc = wmma4(a, b, acc);
    }

    const int   h  = wave;
    const float bb = bias[col0 + l16];
#pragma unroll
    for (int r = 0; r < 8; r++) {
        const int grow = row0 + r + 8 * half;        // global row in (B*T*N)
        const int n  = grow & (Nn - 1);
        const int bt = grow >> 10;                   // /N
        const int t  = bt % T, bbi = bt / T;
        const size_t oi = ((((size_t)(h * Bq + bbi) * T + t) * Nn + n) << 4) + l16;
        OUT[oi] = acc[r] + bb;
    }
}

// ---------------------------------------------------------------------------
// 2) node scores (20x1024 per (hb,t)) via WMMA into LDS, then top-50 per row.
//    softmax skipped: monotone -> same top-k indices. grid=HB*T, block=512.
// ---------------------------------------------------------------------------
__global__ __launch_bounds__(512) void scores_topk(
    const float* __restrict__ qs, const float* __restrict__ ks,
    const float* __restrict__ node_emb, int* __restrict__ idx)
{
    __shared__ float sc[Mm * Nn];                    // 80 KB

    const int bt = blockIdx.x;
    const int hb = bt / T, t = bt % T;
    const float* qb = qs + (size_t)(hb * T + t) * Nn * HD;
    const float* kb = ks + (size_t)(hb * T + t) * Nn * HD;

    const int wave = threadIdx.x >> 5, lane = threadIdx.x & 31;
    const int half = lane >> 4, l16 = lane & 15;

    // 2 row-tiles (m padded to 32) x 64 col-tiles; 16 waves, 8 tiles each
    for (int tile = wave; tile < 128; tile += 16) {
        const int mt = tile >> 6, nt = tile & 63;
        const int m = mt * 16 + l16;
        const int n = nt * 16 + l16;
        v8f acc = {0.f,0.f,0.f,0.f,0.f,0.f,0.f,0.f};
        for (int k = 0; k < 2 * HD; k += 4) {
            const int ka = k + 2 * half;
            v2f a, b;
            a[0] = (m < Mm) ? node_emb[m * (2 * HD) + ka]     : 0.f;
            a[1] = (m < Mm) ? node_emb[m * (2 * HD) + ka + 1] : 0.f;
            b[0] = (ka     < HD) ? qb[(size_t)n * HD + ka]        : kb[(size_t)n * HD + ka - HD];
            b[1] = (ka + 1 < HD) ? qb[(size_t)n * HD + ka + 1]    : kb[(size_t)n * HD + ka + 1 - HD];
            acc = wmma4(a, b, acc);
        }
#pragma unroll
        for (int r = 0; r < 8; r++) {
            const int mm = mt * 16 + r + 8 * half;
            if (mm < Mm) sc[mm * Nn + nt * 16 + l16] = acc[r];
        }
    }
    __syncthreads();

    // top-50 selection: one wave per memory node
    for (int m = wave; m < Mm; m += 16) {
        float v[32];
#pragma unroll
        for (int i = 0; i < 32; i++) v[i] = sc[m * Nn + i * 32 + lane];
        unsigned removed = 0;
        int* ob = idx + ((size_t)(hb * T + t) * Mm + m) * TOPK;
        for (int kk = 0; kk < TOPK; kk++) {
            float bestv = NEG_INF; int bestslot = 0;
#pragma unroll
            for (int i = 0; i < 32; i++) {
                const bool alive = !((removed >> i) & 1u);
                if (alive && v[i] > bestv) { bestv = v[i]; bestslot = i; }
            }
            int bestn = bestslot * 32 + lane;
#pragma unroll
            for (int off = 16; off > 0; off >>= 1) {
                const float ov = __shfl_xor(bestv, off);
                const int   on = __shfl_xor(bestn, off);
                if (ov > bestv || (ov == bestv && on < bestn)) { bestv = ov; bestn = on; }
            }
            if (lane == 0) ob[kk] = bestn;
            if ((bestn & 31) == lane) removed |= (1u << (bestn >> 5));
        }
    }
}

// ---------------------------------------------------------------------------
// 3) per (hb,t,m): gather 50 rows, 64x64-padded WMMA attention, softmax,
//    WMMA @ V, atomic scatter-add. grid=HB*T*M, block=128 (4 waves).
// ---------------------------------------------------------------------------
__global__ __launch_bounds__(128) void node_attn(
    const float* __restrict__ qs, const float* __restrict__ ks,
    const float* __restrict__ vs, const int* __restrict__ idx,
    float* __restrict__ agg, float* __restrict__ cnt)
{
    __shared__ float sq[64 * HD], sk[64 * HD], sv[64 * HD];
    __shared__ float at[64 * 64];
    __shared__ int   sel[TOPK];

    const int gid = blockIdx.x;
    const int m = gid % Mm;
    const int bt = gid / Mm;
    const int t = bt % T, hb = bt / T;
    const size_t nb = (size_t)(hb * T + t) * Nn;
    const float* qb = qs + nb * HD;
    const float* kb = ks + nb * HD;
    const float* vb = vs + nb * HD;
    const int* ib = idx + ((size_t)(hb * T + t) * Mm + m) * TOPK;

    const int tid = threadIdx.x;
    if (tid < TOPK) sel[tid] = ib[tid];
    __syncthreads();

    for (int jj = tid; jj < 64 * HD; jj += 128) {
        const int rr = jj >> 4, e = jj & 15;
        if (rr < TOPK) {
            const int n = sel[rr];
            sq[jj] = qb[(size_t)n * HD + e];
            sk[jj] = kb[(size_t)n * HD + e];
            sv[jj] = vb[(size_t)n * HD + e];
        } else { sq[jj] = 0.f; sk[jj] = 0.f; sv[jj] = 0.f; }
    }
    __syncthreads();

    const int wave = tid >> 5, lane = tid & 31, half = lane >> 4, l16 = lane & 15;
    const int ti = wave;

    // attn = sq @ sk^T * (1/sqrt(16))
    for (int tj = 0; tj < 4; tj++) {
        v8f acc = {0.f,0.f,0.f,0.f,0.f,0.f,0.f,0.f};
#pragma unroll
        for (int k = 0; k < HD; k += 4) {
            const int ka = k + 2 * half;
            v2f a; a[0] = sq[(ti * 16 + l16) * HD + ka]; a[1] = sq[(ti * 16 + l16) * HD + ka + 1];
            v2f b; b[0] = sk[(tj * 16 + l16) * HD + ka]; b[1] = sk[(tj * 16 + l16) * HD + ka + 1];
            acc = wmma4(a, b, acc);
        }
#pragma unroll
        for (int r = 0; r < 8; r++)
            at[(ti * 16 + r + 8 * half) * 64 + tj * 16 + l16] = acc[r] * 0.25f;
    }
    __syncthreads();

    // row softmax over the 50 valid columns; zero the padded columns
    if (tid < 64) {
        float mx = NEG_INF;
        for (int j = 0; j < TOPK; j++) mx = fmaxf(mx, at[tid * 64 + j]);
        float s = 0.f;
        for (int j = 0; j < TOPK; j++) { const float e = __expf(at[tid * 64 + j] - mx); at[tid * 64 + j] = e; s += e; }
        const float inv = 1.0f / s;
        for (int j = 0; j < TOPK; j++) at[tid * 64 + j] *= inv;
        for (int j = TOPK; j < 64; j++) at[tid * 64 + j] = 0.f;
    }
    __syncthreads();

    // node_new = attn @ sv  (M=64 pad, N=16, K=64 pad)
    v8f acc = {0.f,0.f,0.f,0.f,0.f,0.f,0.f,0.f};
#pragma unroll
    for (int k = 0; k < 64; k += 4) {
        const int ka = k + 2 * half;
        v2f a; a[0] = at[(ti * 16 + l16) * 64 + ka]; a[1] = at[(ti * 16 + l16) * 64 + ka + 1];
        v2f b; b[0] = sv[ka * HD + l16];             b[1] = sv[(ka + 1) * HD + l16];
        acc = wmma4(a, b, acc);
    }

    float* ab = agg + nb * HD;
#pragma unroll
    for (int r = 0; r < 8; r++) {
        const int row = ti * 16 + r + 8 * half;
        if (row < TOPK) atomicAdd(&ab[(size_t)sel[row] * HD + l16], acc[r]);
    }
    if (tid < TOPK) atomicAdd(&cnt[nb + sel[tid]], 1.0f);
}

// ---------------------------------------------------------------------------
// 4) normalize by counts + merge heads into (B,T,N,D) row-major (reuses qs)
// ---------------------------------------------------------------------------
__global__ void norm_merge(const float* __restrict__ agg, const float* __restrict__ cnt,
                           float* __restrict__ outm)
{
    const size_t i = (size_t)blockIdx.x * blockDim.x + threadIdx.x;
    if (i >= (size_t)ROWS * D) return;
    const int c = (int)(i & (D - 1));
    const size_t r = i >> 7;
    const int n = (int)(r & (Nn - 1));
    const int bt = (int)(r >> 10);
    const int t = bt % T, bb = bt / T;
    const int h = c >> 4, e = c & 15;
    const size_t base = ((size_t)(h * Bq + bb) * T + t) * Nn + n;
    outm[i] = agg[base * HD + e] / cnt[base];
}

// ---------------------------------------------------------------------------
// 5) output projection: outm @ Wo + bo -> d_out (B,T,N,D)
// ---------------------------------------------------------------------------
__global__ __launch_bounds__(256) void out_proj(
    const float* __restrict__ X, const float* __restrict__ W,
    const float* __restrict__ bias, float* __restrict__ out)
{
    const int wave = threadIdx.x >> 5, lane = threadIdx.x & 31;
    const int half = lane >> 4, l16 = lane & 15;
    const int row0 = blockIdx.x * 16;
    const int col0 = wave * 16;
    const int arow = row0 + l16;

    v8f acc = {0.f,0.f,0.f,0.f,0.f,0.f,0.f,0.f};
    for (int k = 0; k < D; k += 4) {
        const int ka = k + 2 * half;
        if (k + 4 < D) __builtin_prefetch(&W[(k + 4) * D + col0 + l16], 0, 1);
        v2f a; a[0] = X[(size_t)arow * D + ka];       a[1] = X[(size_t)arow * D + ka + 1];
        v2f b; b[0] = W[(size_t)ka * D + col0 + l16]; b[1] = W[(size_t)(ka + 1) * D + col0 + l16];
        acc = wmma4(a, b, acc);
    }
    const float bb = bias[col0 + l16];
#pragma unroll
    for (int r = 0; r < 8; r++) {
        const size_t grow = (size_t)row0 + r + 8 * half;
        out[grow * D + col0 + l16] = acc[r] + bb;
    }
}

// ---------------------------------------------------------------------------
extern "C" void kernel_launch(void* const* d_in, const int* in_sizes, int n_in,
                              void* d_out, int out_size, void* d_ws, size_t ws_size,
                              hipStream_t stream) {
    const float* q_in = (const float*)d_in[0];
    const float* k_in = (const float*)d_in[1];
    const float* v_in = (const float*)d_in[2];
    const float* Wq   = (const float*)d_in[3];
    const float* bq   = (const float*)d_in[4];
    const float* Wk   = (const float*)d_in[5];
    const float* bk   = (const float*)d_in[6];
    const float* Wv   = (const float*)d_in[7];
    const float* bv   = (const float*)d_in[8];
    const float* Wo   = (const float*)d_in[9];
    const float* bo   = (const float*)d_in[10];
    const float* node_emb = (const float*)d_in[11];

    float* qs   = (float*)d_ws;
    float* ksb  = qs  + QN;
    float* vsb  = ksb + QN;
    float* agg  = vsb + QN;
    float* cnt  = agg + QN;
    int*   idxb = (int*)(cnt + CNT_N);

    init_ws<<<(unsigned)((QN + 255) / 256), 256, 0, stream>>>(agg, cnt);

    qkv_proj<<<dim3(ROWS / 16, 3), 256, 0, stream>>>(
        q_in, k_in, v_in, Wq, bq, Wk, bk, Wv, bv, qs, ksb, vsb);

    scores_topk<<<HB * T, 512, 0, stream>>>(qs, ksb, node_emb, idxb);

    node_attn<<<HB * T * Mm, 128, 0, stream>>>(qs, ksb, vsb, idxb, agg, cnt);

    norm_merge<<<(unsigned)(((size_t)ROWS * D + 255) / 256), 256, 0, stream>>>(agg, cnt, qs);

    out_proj<<<dim3(ROWS / 16, 1), 256, 0, stream>>>(qs, Wo, bo, (float*)d_out);
}